// SU4QuantumExtractorModule_29008209117401
// MI455X (gfx1250) — compile-verified
//
#include <hip/hip_runtime.h>
#include <math.h>

// ---------------- problem constants ----------------
#define NQ        9
#define DIM       512          // 2^9
#define NLAYERS   12
#define OBS       256
#define BATCH_TOT 16384

#define WAVES_PER_BLOCK   8     // 8 wave32 = 2 waves per SIMD32 on the WGP
#define STATES_PER_WAVE   8     // 8 states/wave; WMMA rows = (state, r-parity)
#define BLOCK_THREADS     (WAVES_PER_BLOCK * 32)                // 256
#define STATES_PER_BLOCK  (WAVES_PER_BLOCK * STATES_PER_WAVE)   // 64
#define NGATES            (NLAYERS * NQ)                        // 108
#define GATE_FLOATS       128   // 2 K-chunks * 4 k-rows * 16 n-cols (B-fragment layout)
#define GATE_LDS_FLOATS   (NGATES * GATE_FLOATS)                // 13824 floats (55296 B)
#define STATE_LDS_FLOATS  (STATES_PER_WAVE * DIM * 2)           // 8192 floats / wave (32 KB)

typedef float v2f __attribute__((ext_vector_type(2)));
typedef float v8f __attribute__((ext_vector_type(8)));

// ---------------- tiny complex helpers (gate construction) ----------------
struct cpx { float re, im; };
__device__ inline cpx cz() { cpx r; r.re = 0.f; r.im = 0.f; return r; }
__device__ inline cpx cmk(float a, float b) { cpx r; r.re = a; r.im = b; return r; }
__device__ inline cpx cmul(cpx a, cpx b) { return cmk(a.re*b.re - a.im*b.im, a.re*b.im + a.im*b.re); }
__device__ inline cpx cadd(cpx a, cpx b) { return cmk(a.re + b.re, a.im + b.im); }

__device__ inline void u3mat(float t, float p, float l, cpx U[2][2]) {
    float ct = cosf(0.5f * t), st = sinf(0.5f * t);
    float cp = cosf(p), sp = sinf(p), cl = cosf(l), sl = sinf(l);
    U[0][0] = cmk(ct, 0.f);
    U[0][1] = cmk(-cl * st, -sl * st);
    U[1][0] = cmk(cp * st, sp * st);
    U[1][1] = cmk((cp*cl - sp*sl) * ct, (cp*sl + sp*cl) * ct);   // e^{i(p+l)} cos(t/2)
}

__device__ inline void kron22(const cpx A[2][2], const cpx B[2][2], cpx K[4][4]) {
    for (int a = 0; a < 2; ++a)
        for (int b = 0; b < 2; ++b)
            for (int c = 0; c < 2; ++c)
                for (int d = 0; d < 2; ++d)
                    K[2*a + c][2*b + d] = cmul(A[a][b], B[c][d]);
}

__device__ inline void mat4mul(cpx C[4][4], const cpx A[4][4], const cpx B[4][4]) {
    for (int i = 0; i < 4; ++i)
        for (int j = 0; j < 4; ++j) {
            cpx acc = cz();
            for (int k = 0; k < 4; ++k) acc = cadd(acc, cmul(A[i][k], B[k][j]));
            C[i][j] = acc;
        }
}

// Build su4 = post . Rxx.Ryy.Rzz . pre, embed as real 8x8 G8=[[Re,-Im],[Im,Re]],
// store G8^T in the WMMA B-fragment layout with columns 8..15 duplicating 0..7
// (so D cols 8..15 mirror 0..7 -> writeback needs no EXEC masking).
__device__ void computeGate(int g, const float* __restrict__ w, float* __restrict__ gl) {
    const float* th = w + g * 15;
    cpx Ua[2][2], Ub[2][2];
    cpx pre[4][4], post[4][4];
    u3mat(th[0], th[1], th[2], Ua);  u3mat(th[3], th[4], th[5], Ub);  kron22(Ua, Ub, pre);
    u3mat(th[9], th[10], th[11], Ua); u3mat(th[12], th[13], th[14], Ub); kron22(Ua, Ub, post);

    float c6 = cosf(0.5f*th[6]), s6 = sinf(0.5f*th[6]);
    float c7 = cosf(0.5f*th[7]), s7 = sinf(0.5f*th[7]);
    float c8 = cosf(0.5f*th[8]), s8 = sinf(0.5f*th[8]);
    cpx Rxx[4][4], Ryy[4][4], Rzz[4][4];
    for (int i = 0; i < 4; ++i)
        for (int j = 0; j < 4; ++j) { Rxx[i][j] = cz(); Ryy[i][j] = cz(); Rzz[i][j] = cz(); }
    for (int i = 0; i < 4; ++i) { Rxx[i][i] = cmk(c6, 0.f); Ryy[i][i] = cmk(c7, 0.f); }
    Rxx[0][3] = cmk(0.f, -s6); Rxx[1][2] = cmk(0.f, -s6);
    Rxx[2][1] = cmk(0.f, -s6); Rxx[3][0] = cmk(0.f, -s6);
    Ryy[0][3] = cmk(0.f,  s7); Ryy[1][2] = cmk(0.f, -s7);
    Ryy[2][1] = cmk(0.f, -s7); Ryy[3][0] = cmk(0.f,  s7);
    Rzz[0][0] = cmk(c8, -s8); Rzz[1][1] = cmk(c8,  s8);
    Rzz[2][2] = cmk(c8,  s8); Rzz[3][3] = cmk(c8, -s8);

    cpx t1[4][4], t2[4][4];
    mat4mul(t1, Rxx, Ryy);
    mat4mul(t2, t1, Rzz);      // core
    mat4mul(t1, t2, pre);      // core . pre
    mat4mul(t2, post, t1);     // su4

    float* gb = gl + g * GATE_FLOATS;
    for (int c = 0; c < 2; ++c)
        for (int k = 0; k < 4; ++k)
            for (int n = 0; n < 16; ++n) {
                int a = n & 7;                        // duplicate cols 8..15 = cols 0..7
                int b = 4*c + k;                      // B[kglob][n] = G8[n&7][kglob]
                float val;
                if (a < 4) val = (b < 4) ?  t2[a][b].re     : -t2[a][b-4].im;
                else       val = (b < 4) ?  t2[a-4][b].im   :  t2[a-4][b-4].re;
                gb[c*64 + k*16 + n] = val;
            }
}

// insert two bits (positions lo<hi with values vlo,vhi) into 7-bit r -> 9-bit amp index
__device__ inline unsigned insert2(unsigned r, int lo, unsigned vlo, int hi, unsigned vhi) {
    unsigned low = r & ((1u << lo) - 1u);
    unsigned x = ((r >> lo) << (lo + 1)) | (vlo << lo) | low;
    unsigned low2 = x & ((1u << hi) - 1u);
    return ((x >> hi) << (hi + 1)) | (vhi << hi) | low2;
}
// pm = bit position carrying (p>>1), pl = bit position carrying (p&1)
__device__ inline unsigned ampIndex(unsigned r, int pm, int pl, unsigned p) {
    unsigned vm = (p >> 1) & 1u, vl = p & 1u;
    int lo  = (pm < pl) ? pm : pl;  unsigned vlo = (pm < pl) ? vm : vl;
    int hi  = (pm < pl) ? pl : pm;  unsigned vhi = (pm < pl) ? vl : vm;
    return insert2(r, lo, vlo, hi, vhi);
}

__global__ __launch_bounds__(BLOCK_THREADS, 1)
void su4_quantum_kernel(const float* __restrict__ features,
                        const float* __restrict__ weights,
                        float* __restrict__ out) {
    extern __shared__ float smem[];
    float* gl = smem;                                     // 108 gate B-fragments
    const int tid  = threadIdx.x;
    const int lane = tid & 31;
    const int wave = tid >> 5;
    float*  stateF = smem + GATE_LDS_FLOATS + wave * STATE_LDS_FLOATS;
    float2* stateV = (float2*)stateF;

    // ---- phase 1: all 108 gate matrices (redundant per block; ~100 KFLOP) ----
    if (tid < NGATES) computeGate(tid, weights, gl);

    // ---- phase 2: load + normalize features into this wave's LDS states ----
    const int bBase = (blockIdx.x * WAVES_PER_BLOCK + wave) * STATES_PER_WAVE;
    for (int s = 0; s < STATES_PER_WAVE; ++s) {
        const float* f = features + (size_t)(bBase + s) * OBS;
        float ss = 0.f;
        for (int j = lane; j < OBS; j += 32) { float v = f[j]; ss += v * v; }
        for (int msk = 16; msk >= 1; msk >>= 1) ss += __shfl_xor(ss, msk, 32);
        float inv = 1.0f / (sqrtf(ss) + 1e-8f);
        for (int j = lane; j < DIM; j += 32) {
            float re = (j < OBS) ? f[j] * inv : 0.f;
            stateV[s * DIM + j] = make_float2(re, 0.f);
        }
    }
    __syncthreads();

    // ---- phase 3: 108 sequential gates, state resident in LDS ----
    // WMMA row m = (state s = m&7, r-parity rsub = m>>3); r = 2*rr + rsub.
    const int s    = lane & 7;                          // A-row state
    const unsigned rsub = (unsigned)((lane >> 3) & 1);  // A-row r parity
    const int half = lane >> 4;           // selects K pair (A) / k rows (B); D row half
    const int kb   = 2 * half;
    const int ncol = lane & 15;           // B/D column
    const unsigned q = (unsigned)(ncol & 3);            // output amp-pair index
    const int reim = (ncol >> 2) & 1;                   // 0 = re, 1 = im

    for (int layer = 0; layer < NLAYERS; ++layer) {
        for (int i = 0; i < NQ; ++i) {
            const int g = layer * NQ + i;
            const float* gb = gl + g * GATE_FLOATS;
            v2f B0, B1;                               // K rows 0..3 (re-part), 4..7 (im-part)
            B0[0] = gb[(kb    ) * 16 + ncol];
            B0[1] = gb[(kb + 1) * 16 + ncol];
            B1[0] = gb[64 + (kb    ) * 16 + ncol];
            B1[1] = gb[64 + (kb + 1) * 16 + ncol];
            const int pm = (i < 8) ? (8 - i) : 0;     // bit holding p>>1 (qubit i)
            const int pl = (i < 8) ? (7 - i) : 8;     // bit holding p&1  (qubit i+1 mod 9)

            // software pipeline: x = A-data for iteration rr, y = prefetch for rr+1
            float2 x0 = stateV[s * DIM + ampIndex(rsub, pm, pl, (unsigned)kb)];
            float2 x1 = stateV[s * DIM + ampIndex(rsub, pm, pl, (unsigned)(kb + 1))];

            for (int rr = 0; rr < 64; ++rr) {
                unsigned rn = 2u * (unsigned)((rr + 1) & 63) + rsub;   // prefetch (wraps harmlessly)
                float2 y0 = stateV[s * DIM + ampIndex(rn, pm, pl, (unsigned)kb)];
                float2 y1 = stateV[s * DIM + ampIndex(rn, pm, pl, (unsigned)(kb + 1))];

                v2f A0; A0[0] = x0.x; A0[1] = x1.x;   // real parts  (K 0..3)
                v2f A1; A1[0] = x0.y; A1[1] = x1.y;   // imag parts  (K 4..7)
                v8f acc = {0.f, 0.f, 0.f, 0.f, 0.f, 0.f, 0.f, 0.f};
                acc = __builtin_amdgcn_wmma_f32_16x16x4_f32(false, A0, false, B0,
                                                            (short)0, acc, false, false);
                acc = __builtin_amdgcn_wmma_f32_16x16x4_f32(false, A1, false, B1,
                                                            (short)0, acc, false, false);

                // Unconditional writeback: D cols 8..15 duplicate 0..7, so lane pairs
                // (L, L^8) store identical values to identical LDS addresses.
                // D rows: M = v + 8*half -> state v, r-parity = half (lane-uniform).
                unsigned aw = ampIndex(2u * (unsigned)rr + (unsigned)half, pm, pl, q);
                #pragma unroll
                for (int v = 0; v < 8; ++v)
                    stateF[v * (DIM * 2) + aw * 2 + reim] = acc[v];

                x0 = y0; x1 = y1;
            }
        }
    }

    // ---- phase 4: probs -> tanh(5p) to global (wave-private LDS, in-order) ----
    for (int ss2 = 0; ss2 < STATES_PER_WAVE; ++ss2) {
        float* o = out + (size_t)(bBase + ss2) * DIM;
        for (int j = lane; j < DIM; j += 32) {
            float2 a = stateV[ss2 * DIM + j];
            o[j] = tanhf(5.0f * (a.x * a.x + a.y * a.y));
        }
    }
}

extern "C" void kernel_launch(void* const* d_in, const int* in_sizes, int n_in,
                              void* d_out, int out_size, void* d_ws, size_t ws_size,
                              hipStream_t stream) {
    (void)in_sizes; (void)n_in; (void)d_ws; (void)ws_size; (void)out_size;
    const float* features = (const float*)d_in[0];   // (16384, 256) f32
    const float* weights  = (const float*)d_in[1];   // (12, 9, 15)  f32
    float* out = (float*)d_out;                      // (16384, 512) f32

    dim3 grid(BATCH_TOT / STATES_PER_BLOCK);         // 256 workgroups
    dim3 block(BLOCK_THREADS);                       // 256 threads = 8 wave32
    size_t shmem = (size_t)(GATE_LDS_FLOATS + WAVES_PER_BLOCK * STATE_LDS_FLOATS)
                   * sizeof(float);                  // 317440 B (< 320 KB/WGP)
    su4_quantum_kernel<<<grid, block, shmem, stream>>>(features, weights, out);
}